// CRFDecoder_40398462386765
// MI455X (gfx1250) — compile-verified
//
#include <hip/hip_runtime.h>
#include <hip/hip_bf16.h>
#include <math.h>

// Problem constants (from reference): S=512, B=256, H=512, V=256
#define CRF_S 512
#define CRF_B 256
#define CRF_H 512
#define CRF_V 256

typedef float v2f __attribute__((ext_vector_type(2)));
typedef float v8f __attribute__((ext_vector_type(8)));

// D = A(16x4, f32) * B(4x16, f32) + C(16x16, f32)
__device__ __forceinline__ v8f wmma4(v2f a, v2f b, v8f c) {
    return __builtin_amdgcn_wmma_f32_16x16x4_f32(
        /*neg_a=*/false, a, /*neg_b=*/false, b,
        /*c_mod=*/(short)0, c, /*reuse_a=*/false, /*reuse_b=*/false);
}

// -------------------------------------------------------------------------
// K1: per-column max of transition + expT[i][j] = exp(T[i][j] - cmax[j])
// One block per column j, 256 threads (one per row i).
// -------------------------------------------------------------------------
__global__ __launch_bounds__(256) void k_prep(const float* __restrict__ T,
                                              float* __restrict__ expT,
                                              float* __restrict__ cmax) {
    __shared__ float red[256];
    const int j = blockIdx.x;
    const int i = threadIdx.x;
    const float v = T[i * CRF_V + j];
    red[i] = v;
    __syncthreads();
    for (int off = 128; off > 0; off >>= 1) {
        if (i < off) red[i] = fmaxf(red[i], red[i + off]);
        __syncthreads();
    }
    const float c = red[0];
    expT[i * CRF_V + j] = __expf(v - c);
    if (i == 0) cmax[j] = c;
}

// -------------------------------------------------------------------------
// K2: logits = log_softmax(enc_outs @ W + b), fused.
// Block = 256 threads (8 waves). Each block: 16 rows of flattened (S*B, H)
// times full V=256. Wave w owns N-tiles {2w, 2w+1}. K = H = 512, step 4
// via V_WMMA_F32_16X16X4_F32. log_softmax reduced across the block in LDS.
// -------------------------------------------------------------------------
__global__ __launch_bounds__(256) void k_logits(const float* __restrict__ X,
                                                const float* __restrict__ W,
                                                const float* __restrict__ bias,
                                                float* __restrict__ logits) {
    __shared__ float At[16][516];     // A tile 16 x 512, padded (516%64==4)
    __shared__ float Ct[16][260];     // C tile 16 x 256, padded
    __shared__ float red[16][16];
    __shared__ float rowmax[16];
    __shared__ float rowlse[16];

    const int t = threadIdx.x;
    const int lane = t & 31;
    const int wave = t >> 5;
    const size_t row0 = (size_t)blockIdx.x * 16;

    // Stage A tile: 16 rows x 512 f32 = 2048 float4's, 8 per thread.
    for (int i = t; i < 2048; i += 256) {
        const int r = i >> 7;        // 128 float4 per row
        const int c4 = i & 127;
        const float4 v = ((const float4*)(X + (row0 + r) * CRF_H))[c4];
        At[r][c4 * 4 + 0] = v.x; At[r][c4 * 4 + 1] = v.y;
        At[r][c4 * 4 + 2] = v.z; At[r][c4 * 4 + 3] = v.w;
    }
    __syncthreads();

    // Fragment coordinates (ISA 7.12.2, 32-bit A 16x4 / C 16x16 layouts)
    const int m  = lane & 15;
    const int ko = (lane >> 4) << 1;       // 0 or 2
    const int n0 = wave * 32 + m;          // tile 0 column
    const int n1 = n0 + 16;                // tile 1 column

    v8f acc0 = {0.f,0.f,0.f,0.f,0.f,0.f,0.f,0.f};
    v8f acc1 = {0.f,0.f,0.f,0.f,0.f,0.f,0.f,0.f};

    for (int k0 = 0; k0 < CRF_H; k0 += 4) {
        v2f a;  a.x  = At[m][k0 + ko];               a.y  = At[m][k0 + ko + 1];
        v2f b0; b0.x = W[(k0 + ko) * CRF_V + n0];    b0.y = W[(k0 + ko + 1) * CRF_V + n0];
        v2f b1; b1.x = W[(k0 + ko) * CRF_V + n1];    b1.y = W[(k0 + ko + 1) * CRF_V + n1];
        acc0 = wmma4(a, b0, acc0);
        acc1 = wmma4(a, b1, acc1);
    }

    const float bv0 = bias[n0];
    const float bv1 = bias[n1];
    #pragma unroll
    for (int r = 0; r < 8; ++r) {
        const int mr = (lane < 16) ? r : (r + 8);
        Ct[mr][n0] = acc0[r] + bv0;
        Ct[mr][n1] = acc1[r] + bv1;
    }
    __syncthreads();

    // log_softmax over V=256 per row; thread -> (row = t&15, chunk = t>>4)
    const int row = t & 15;
    const int chunk = t >> 4;
    float pm = -3.4e38f;
    #pragma unroll
    for (int j = 0; j < 16; ++j) pm = fmaxf(pm, Ct[row][chunk * 16 + j]);
    red[row][chunk] = pm;
    __syncthreads();
    if (t < 16) {
        float mm = red[t][0];
        for (int c = 1; c < 16; ++c) mm = fmaxf(mm, red[t][c]);
        rowmax[t] = mm;
    }
    __syncthreads();
    const float rm = rowmax[row];
    float ps = 0.f;
    #pragma unroll
    for (int j = 0; j < 16; ++j) ps += __expf(Ct[row][chunk * 16 + j] - rm);
    red[row][chunk] = ps;
    __syncthreads();
    if (t < 16) {
        float ss = 0.f;
        for (int c = 0; c < 16; ++c) ss += red[t][c];
        rowlse[t] = __logf(ss);
    }
    __syncthreads();
    const float off = rm + rowlse[row];
    float* dst = logits + (row0 + row) * CRF_V;
    #pragma unroll
    for (int j = 0; j < 16; ++j)
        dst[chunk * 16 + j] = Ct[row][chunk * 16 + j] - off;
}

// -------------------------------------------------------------------------
// K3: CRF forward scan, exp-space matmul formulation, persistent blocks.
// Grid = 16 blocks (16 batch rows each), 256 threads. expT (256 KB), Z and
// p tiles all LDS-resident (~292 KB of the WGP's 320 KB).
// Per step: m_b = max_j Z ; p = exp(Z - m) ; C = p @ expT (WMMA) ;
//           Zn = log(C) + m_b + cmax_j + logits[s] ; mask-select.
// -------------------------------------------------------------------------
__global__ __launch_bounds__(256) void k_scan(const float* __restrict__ logits,
                                              const float* __restrict__ expT,
                                              const float* __restrict__ cmax,
                                              const float* __restrict__ startT,
                                              const float* __restrict__ endT,
                                              const int*   __restrict__ lengths,
                                              float* __restrict__ Zp) {
    __shared__ float eT[CRF_V * CRF_V];   // 256 KB
    __shared__ float Zb[16][256];         // 16 KB
    __shared__ float Pb[16][264];         // p / C tile (reused), padded
    __shared__ float red[16][16];
    __shared__ float mrow[16];
    __shared__ float cj[256];
    __shared__ float ej[256];
    __shared__ int   lenb[16];

    const int t = threadIdx.x;
    const int lane = t & 31;
    const int wave = t >> 5;
    const int bb0 = blockIdx.x * 16;

    // Stage expT (64 float4 per thread)
    for (int i = t; i < 16384; i += 256) {
        const float4 v = ((const float4*)expT)[i];
        eT[i * 4 + 0] = v.x; eT[i * 4 + 1] = v.y;
        eT[i * 4 + 2] = v.z; eT[i * 4 + 3] = v.w;
    }
    cj[t] = cmax[t];
    ej[t] = endT[t];
    if (t < 16) lenb[t] = lengths[bb0 + t];

    const int row = t & 15;
    const int chunk = t >> 4;
    // Z0 = start_transition + logits[0]
    #pragma unroll
    for (int j = 0; j < 16; ++j) {
        const int col = chunk * 16 + j;
        Zb[row][col] = startT[col] + logits[(size_t)(bb0 + row) * CRF_V + col];
    }
    __syncthreads();

    const int m  = lane & 15;
    const int ko = (lane >> 4) << 1;
    const int n0 = wave * 32 + m;
    const int n1 = n0 + 16;

    for (int s = 1; s < CRF_S; ++s) {
        // 1) per-row max of Z
        float pm = -3.4e38f;
        #pragma unroll
        for (int j = 0; j < 16; ++j) pm = fmaxf(pm, Zb[row][chunk * 16 + j]);
        red[row][chunk] = pm;
        __syncthreads();
        if (t < 16) {
            float mm = red[t][0];
            for (int c = 1; c < 16; ++c) mm = fmaxf(mm, red[t][c]);
            mrow[t] = mm;
        }
        __syncthreads();
        const float rm = mrow[row];
        // 2) p = exp(Z - m)
        #pragma unroll
        for (int j = 0; j < 16; ++j) {
            const int col = chunk * 16 + j;
            Pb[row][col] = __expf(Zb[row][col] - rm);
        }
        __syncthreads();
        // 3) C = p (16x256) @ expT (256x256), WMMA f32 16x16x4
        v8f acc0 = {0.f,0.f,0.f,0.f,0.f,0.f,0.f,0.f};
        v8f acc1 = {0.f,0.f,0.f,0.f,0.f,0.f,0.f,0.f};
        for (int k0 = 0; k0 < CRF_V; k0 += 4) {
            v2f a;  a.x  = Pb[m][k0 + ko];               a.y  = Pb[m][k0 + ko + 1];
            v2f b0; b0.x = eT[(k0 + ko) * CRF_V + n0];   b0.y = eT[(k0 + ko + 1) * CRF_V + n0];
            v2f b1; b1.x = eT[(k0 + ko) * CRF_V + n1];   b1.y = eT[(k0 + ko + 1) * CRF_V + n1];
            acc0 = wmma4(a, b0, acc0);
            acc1 = wmma4(a, b1, acc1);
        }
        __syncthreads();   // all waves done reading Pb
        #pragma unroll
        for (int r = 0; r < 8; ++r) {
            const int mr = (lane < 16) ? r : (r + 8);
            Pb[mr][n0] = acc0[r];
            Pb[mr][n1] = acc1[r];
        }
        __syncthreads();
        // 4) Zn = log(C) + m + cmax_j + logit ; masked update
        const bool upd = (s < lenb[row]);
        const size_t lbase = ((size_t)s * CRF_B + bb0 + row) * CRF_V;
        #pragma unroll
        for (int j = 0; j < 16; ++j) {
            const int col = chunk * 16 + j;
            const float zn = __logf(Pb[row][col]) + rm + cj[col] + logits[lbase + col];
            if (upd) Zb[row][col] = zn;
        }
        __syncthreads();
    }

    // Zp = logsumexp(Zf + end_transition)
    float pm = -3.4e38f;
    #pragma unroll
    for (int j = 0; j < 16; ++j) {
        const int col = chunk * 16 + j;
        pm = fmaxf(pm, Zb[row][col] + ej[col]);
    }
    red[row][chunk] = pm;
    __syncthreads();
    if (t < 16) {
        float mm = red[t][0];
        for (int c = 1; c < 16; ++c) mm = fmaxf(mm, red[t][c]);
        mrow[t] = mm;
    }
    __syncthreads();
    const float rm2 = mrow[row];
    float ps = 0.f;
    #pragma unroll
    for (int j = 0; j < 16; ++j) {
        const int col = chunk * 16 + j;
        ps += __expf(Zb[row][col] + ej[col] - rm2);
    }
    red[row][chunk] = ps;
    __syncthreads();
    if (t < 16) {
        float ss = 0.f;
        for (int c = 0; c < 16; ++c) ss += red[t][c];
        Zp[bb0 + t] = mrow[t] + __logf(ss);
    }
}

// -------------------------------------------------------------------------
// K4: gold-path score per batch element + mean(Zp - score) -> d_out[0].
// Single block, thread b handles batch element b (coalesced over b).
// -------------------------------------------------------------------------
__global__ __launch_bounds__(256) void k_score(const float* __restrict__ logits,
                                               const float* __restrict__ T,
                                               const float* __restrict__ startT,
                                               const float* __restrict__ endT,
                                               const int*   __restrict__ targets,
                                               const int*   __restrict__ lengths,
                                               const float* __restrict__ Zp,
                                               float* __restrict__ out) {
    __shared__ float red[256];
    const int b = threadIdx.x;
    const int len = lengths[b];
    int cur = targets[b];                 // targets[0][b]
    float sc = startT[cur];
    for (int s = 0; s < CRF_S - 1; ++s) {
        const int nxt = targets[(s + 1) * CRF_B + b];
        const float mnext = ((s + 1) < len) ? 1.f : 0.f;
        const float mcur  = (s < len) ? 1.f : 0.f;
        sc += T[cur * CRF_V + nxt] * mnext
            + logits[((size_t)s * CRF_B + b) * CRF_V + cur] * mcur;
        cur = nxt;
    }
    const int lt = targets[(len - 1) * CRF_B + b];
    const float mlast = ((CRF_S - 1) < len) ? 1.f : 0.f;
    sc += endT[lt] + logits[((size_t)(CRF_S - 1) * CRF_B + b) * CRF_V + lt] * mlast;

    red[b] = Zp[b] - sc;
    __syncthreads();
    for (int off = 128; off > 0; off >>= 1) {
        if (b < off) red[b] += red[b + off];
        __syncthreads();
    }
    if (b == 0) out[0] = red[0] / (float)CRF_B;
}

// -------------------------------------------------------------------------
extern "C" void kernel_launch(void* const* d_in, const int* in_sizes, int n_in,
                              void* d_out, int out_size, void* d_ws, size_t ws_size,
                              hipStream_t stream) {
    const float* enc     = (const float*)d_in[0];  // (S,B,H)
    const float* W       = (const float*)d_in[1];  // (H,V)
    const float* bias    = (const float*)d_in[2];  // (V)
    const float* T       = (const float*)d_in[3];  // (V,V)
    const float* startT  = (const float*)d_in[4];  // (V)
    const float* endT    = (const float*)d_in[5];  // (V)
    const int*   targets = (const int*)d_in[6];    // (S,B)
    const int*   lengths = (const int*)d_in[7];    // (B)

    float* ws     = (float*)d_ws;
    float* logits = ws;                                       // S*B*V = 33554432 f32
    float* expT   = ws + (size_t)CRF_S * CRF_B * CRF_V;       // V*V   = 65536 f32
    float* cmax   = expT + (size_t)CRF_V * CRF_V;             // V
    float* Zp     = cmax + CRF_V;                             // B

    k_prep  <<<CRF_V, 256, 0, stream>>>(T, expT, cmax);
    k_logits<<<(CRF_S * CRF_B) / 16, 256, 0, stream>>>(enc, W, bias, logits);
    k_scan  <<<CRF_B / 16, 256, 0, stream>>>(logits, expT, cmax, startT, endT, lengths, Zp);
    k_score <<<1, 256, 0, stream>>>(logits, T, startT, endT, targets, lengths, Zp, (float*)d_out);
}